// RoutingCapsule_20212116095124
// MI455X (gfx1250) — compile-verified
//
#include <hip/hip_runtime.h>
#include <hip/hip_bf16.h>

typedef __attribute__((ext_vector_type(2))) float    v2f;
typedef __attribute__((ext_vector_type(8))) float    v8f;
typedef __attribute__((ext_vector_type(8))) _Float16 v8h;
typedef __attribute__((ext_vector_type(4))) int      v4i;

#define N_IN   2048
#define N_OUT  64
#define D_OUT  32
#define D_IN   16
#define BATCH  32
#define JD     (N_OUT * D_OUT)   // 2048
#define IB     128               // capsules per routing workgroup

// ---------------------------------------------------------------------------
// CDNA5 async global->LDS copy (GLOBAL_LOAD_ASYNC_TO_LDS_B128, ASYNCcnt).
// Builtin params are (v4i addrspace(1)*, v4i addrspace(3)*, imm offset, imm cpol).
// ---------------------------------------------------------------------------
#if defined(__AMDGCN__) && __has_builtin(__builtin_amdgcn_global_load_async_to_lds_b128)
#define HAVE_ASYNC_LDS 1
typedef __attribute__((address_space(1))) v4i gv4i;
typedef __attribute__((address_space(3))) v4i lv4i;
__device__ __forceinline__ void async_b128(const void* g, void* l) {
  __builtin_amdgcn_global_load_async_to_lds_b128((gv4i*)g, (lv4i*)l, 0, 0);
}
template <int N>
__device__ __forceinline__ void wait_async() {
  asm volatile("s_wait_asynccnt %0" ::"i"(N) : "memory");
}
#else
#define HAVE_ASYNC_LDS 0
#endif

// ---------------------------------------------------------------------------
// Kernel 1: u_hat[b,i,jd] = sum_p x[b,i,p] * W[i,jd,p]
// One WG per capsule i; 8 waves; per wave: 2 M-tiles (b) x 16 N-tiles (jd)
// x 4 k-steps of V_WMMA_F32_16X16X4_F32.  W streamed via double-buffered
// async global->LDS b128 copies; fp16 output staged in LDS for b128 stores.
// ---------------------------------------------------------------------------
__global__ __launch_bounds__(256) void gemm_uhat_kernel(
    const float* __restrict__ x, const float* __restrict__ W,
    _Float16* __restrict__ uhat) {
  const int i    = blockIdx.x;        // capsule 0..2047
  const int t    = threadIdx.x;
  const int lane = t & 31;
  const int wave = t >> 5;            // 0..7
  const int hl   = lane >> 4;         // half-wave (K-pair select)
  const int l16  = lane & 15;

  __shared__ float xs[BATCH][D_IN];              // 2 KB
#if HAVE_ASYNC_LDS
  __shared__ __align__(16) float wbuf[2][8][16 * 20];   // 20 KB, 80B padded rows
#endif
  __shared__ __align__(16) _Float16 ostage[8][32][24];  // 12 KB, 48B rows

  for (int e = t; e < BATCH * D_IN; e += 256) {
    int b = e >> 4, p = e & 15;
    xs[b][p] = x[((size_t)b * N_IN + i) * D_IN + p];
  }
  __syncthreads();

  // A fragments (f32 A 16x4): lanes 0-15 M with K={0,1}; lanes 16-31 K={2,3}
  v2f afrag[2][4];
#pragma unroll
  for (int mt = 0; mt < 2; ++mt)
#pragma unroll
    for (int kk = 0; kk < 4; ++kk) {
      const int p0 = 4 * kk + 2 * hl;
      afrag[mt][kk].x = xs[mt * 16 + l16][p0];
      afrag[mt][kk].y = xs[mt * 16 + l16][p0 + 1];
    }

  const float* Wi = W + (size_t)i * JD * D_IN;

#if HAVE_ASYNC_LDS
  // wave-private 1KB chunk per round: rows jd0..jd0+15 (16 floats each).
  // lane L copies chunk floats [8L, 8L+8) -> LDS row L/2, col (L&1)*8.
  auto issueW = [&](int nt) {
    const int    jd0 = wave * 256 + nt * 16;
    const float* src = Wi + (size_t)jd0 * D_IN + lane * 8;
    float*       dst = &wbuf[nt & 1][wave][(lane >> 1) * 20 + (lane & 1) * 8];
    async_b128(src, dst);
    async_b128(src + 4, dst + 4);
  };
  issueW(0);
#endif

  for (int nt = 0; nt < 16; ++nt) {
    const int jd0 = wave * 256 + nt * 16;

    v2f bfrag[4];
#if HAVE_ASYNC_LDS
    if (nt + 1 < 16) { issueW(nt + 1); wait_async<2>(); }
    else             { wait_async<0>(); }
    const float* wr = &wbuf[nt & 1][wave][l16 * 20 + 2 * hl];
#pragma unroll
    for (int kk = 0; kk < 4; ++kk) {
      bfrag[kk].x = wr[4 * kk];
      bfrag[kk].y = wr[4 * kk + 1];
    }
#else
    const float* wrow = Wi + (size_t)(jd0 + l16) * D_IN + 2 * hl;
    if (nt + 1 < 16) __builtin_prefetch(wrow + 16 * D_IN, 0, 1);
#pragma unroll
    for (int kk = 0; kk < 4; ++kk) {
      bfrag[kk].x = wrow[4 * kk];
      bfrag[kk].y = wrow[4 * kk + 1];
    }
#endif

    v8f acc0 = {}, acc1 = {};
#pragma unroll
    for (int kk = 0; kk < 4; ++kk) {
      acc0 = __builtin_amdgcn_wmma_f32_16x16x4_f32(
          false, afrag[0][kk], false, bfrag[kk], (short)0, acc0, false, false);
      acc1 = __builtin_amdgcn_wmma_f32_16x16x4_f32(
          false, afrag[1][kk], false, bfrag[kk], (short)0, acc1, false, false);
    }

    // Stage fp16 tile in LDS (wave-internal, DS ops in-order: no barrier),
    // then emit coalesced b128 stores: lane = batch row.
#pragma unroll
    for (int r = 0; r < 8; ++r) {
      ostage[wave][r + 8 * hl][l16]      = (_Float16)acc0[r];
      ostage[wave][16 + r + 8 * hl][l16] = (_Float16)acc1[r];
    }
    const v8h s0 = *(const v8h*)&ostage[wave][lane][0];
    const v8h s1 = *(const v8h*)&ostage[wave][lane][8];
    _Float16* orow = uhat + ((size_t)lane * N_IN + i) * JD + jd0;
    *(v8h*)(orow)     = s0;
    *(v8h*)(orow + 8) = s1;
  }
}

// ---------------------------------------------------------------------------
// Routing pass. One WG = (b, block of IB capsules). Streams u_hat once via
// double-buffered async global->LDS copies.
// mode 0: c uniform (iter 0)                 -> accumulate s0
// mode 1: a = u.v;  blog = a; c = softmax(a) -> accumulate s1
// mode 2: a = blog + u.v;     c = softmax(a) -> accumulate s2
// ---------------------------------------------------------------------------
__global__ __launch_bounds__(256) void routing_pass_kernel(
    const _Float16* __restrict__ uhat, const float* __restrict__ vin,
    float* __restrict__ blog, float* __restrict__ sout, int mode) {
  const int b  = blockIdx.y;
  const int i0 = blockIdx.x * IB;
  const int t  = threadIdx.x;
  const int j  = t >> 2;              // output capsule handled by lane group

  __shared__ float vsh[JD];           // 8 KB
  __shared__ float spart[JD];         // 8 KB
  __shared__ float ash[N_OUT];
  __shared__ float esh[N_OUT];
#if HAVE_ASYNC_LDS
  __shared__ __align__(16) _Float16 ubuf[2][JD];  // 8 KB double buffer
#endif

#pragma unroll
  for (int e = 0; e < 8; ++e) spart[t * 8 + e] = 0.f;
  if (mode > 0) {
#pragma unroll
    for (int e = 0; e < 8; ++e)
      vsh[t + 256 * e] = vin[(size_t)b * JD + t + 256 * e];
  }
  __syncthreads();

  const _Float16* ubase = uhat + ((size_t)b * N_IN + i0) * JD;
#if HAVE_ASYNC_LDS
  // thread t owns jd = 8t..8t+7 (16B): async-copy its slice of row ii.
  async_b128(ubase + t * 8, &ubuf[0][t * 8]);
#endif

  for (int ii = 0; ii < IB; ++ii) {
    const int i = i0 + ii;
    v8h uh;
#if HAVE_ASYNC_LDS
    if (ii + 1 < IB) {
      async_b128(ubase + (size_t)(ii + 1) * JD + t * 8, &ubuf[(ii + 1) & 1][t * 8]);
      wait_async<1>();
    } else {
      wait_async<0>();
    }
    uh = *(const v8h*)&ubuf[ii & 1][t * 8];
#else
    uh = *((const v8h*)(ubase + (size_t)ii * JD) + t);
#endif
    float u[8];
#pragma unroll
    for (int e = 0; e < 8; ++e) u[e] = (float)uh[e];

    float c;
    if (mode > 0) {
      // agreement: 8 local MACs + reduce across the 4-lane group sharing j
      float dot = 0.f;
#pragma unroll
      for (int e = 0; e < 8; ++e) dot += u[e] * vsh[t * 8 + e];
      dot += __shfl_xor(dot, 1, 32);
      dot += __shfl_xor(dot, 2, 32);
      float a = dot;
      const size_t bidx = ((size_t)b * N_IN + i) * N_OUT + j;
      if (mode == 2) a += blog[bidx];
      if ((t & 3) == 0) {
        if (mode == 1) blog[bidx] = a;
        ash[j] = a;
      }
      __syncthreads();
      float m = -1e30f;
#pragma unroll 16
      for (int q = 0; q < N_OUT; ++q) m = fmaxf(m, ash[q]);
      if (t < N_OUT) esh[t] = __expf(ash[t] - m);
      __syncthreads();
      float sum = 0.f;
#pragma unroll 16
      for (int q = 0; q < N_OUT; ++q) sum += esh[q];
      c = esh[j] / sum;
      __syncthreads();  // protect ash/esh before next capsule
    } else {
      c = 1.0f / 64.0f;
    }
#pragma unroll
    for (int e = 0; e < 8; ++e) spart[t * 8 + e] += c * u[e];
  }
  __syncthreads();
#pragma unroll
  for (int e = 0; e < 8; ++e)
    atomicAdd(&sout[(size_t)b * JD + t * 8 + e], spart[t * 8 + e]);
}

// ---------------------------------------------------------------------------
// squash(s + bias) -> v.  One wave per (b,j); full-wave reduction for |s|^2.
// ---------------------------------------------------------------------------
__global__ __launch_bounds__(32) void squash_kernel(
    const float* __restrict__ s, const float* __restrict__ bias,
    float* __restrict__ vout) {
  const int bj = blockIdx.x;          // b*64 + j
  const int j  = bj & (N_OUT - 1);
  const int d  = threadIdx.x;
  float sv = s[(size_t)bj * D_OUT + d] + bias[j * D_OUT + d];
  float sq = sv * sv;
#pragma unroll
  for (int off = 16; off > 0; off >>= 1) sq += __shfl_xor(sq, off, 32);
  const float scale = sq / ((1.f + sq) * sqrtf(sq + 1e-7f));
  vout[(size_t)bj * D_OUT + d] = sv * scale;
}

__global__ void zero_kernel(float* __restrict__ p, int n) {
  const int idx = blockIdx.x * blockDim.x + threadIdx.x;
  if (idx < n) p[idx] = 0.f;
}

// ---------------------------------------------------------------------------
extern "C" void kernel_launch(void* const* d_in, const int* in_sizes, int n_in,
                              void* d_out, int out_size, void* d_ws, size_t ws_size,
                              hipStream_t stream) {
  const float* x    = (const float*)d_in[0];   // (32,2048,16)
  const float* W    = (const float*)d_in[1];   // (2048,64,32,16)
  const float* bias = (const float*)d_in[2];   // (64,32)
  float* out = (float*)d_out;                  // (32,64,32)

  char* ws = (char*)d_ws;
  _Float16* uhat = (_Float16*)ws;                           // 256 MB
  size_t off = (size_t)BATCH * N_IN * JD * sizeof(_Float16);
  float* blog = (float*)(ws + off);                         // 16 MB
  off += (size_t)BATCH * N_IN * N_OUT * sizeof(float);
  float* s = (float*)(ws + off);                            // 256 KB
  off += (size_t)BATCH * JD * sizeof(float);
  float* v = (float*)(ws + off);                            // 256 KB

  const int sN = BATCH * JD;                                // 65536
  const dim3 rgrid(N_IN / IB, BATCH);                       // (16, 32)

  gemm_uhat_kernel<<<N_IN, 256, 0, stream>>>(x, W, uhat);

  // iter 0: uniform coupling
  zero_kernel<<<(sN + 255) / 256, 256, 0, stream>>>(s, sN);
  routing_pass_kernel<<<rgrid, 256, 0, stream>>>(uhat, v, blog, s, 0);
  squash_kernel<<<BATCH * N_OUT, 32, 0, stream>>>(s, bias, v);

  // iter 1: b = u.v0
  zero_kernel<<<(sN + 255) / 256, 256, 0, stream>>>(s, sN);
  routing_pass_kernel<<<rgrid, 256, 0, stream>>>(uhat, v, blog, s, 1);
  squash_kernel<<<BATCH * N_OUT, 32, 0, stream>>>(s, bias, v);

  // iter 2: b += u.v1 (no write-back needed)
  zero_kernel<<<(sN + 255) / 256, 256, 0, stream>>>(s, sN);
  routing_pass_kernel<<<rgrid, 256, 0, stream>>>(uhat, v, blog, s, 2);
  squash_kernel<<<BATCH * N_OUT, 32, 0, stream>>>(s, bias, out);
}